// EqualizeClahe_55327768708740
// MI455X (gfx1250) — compile-verified
//
#include <hip/hip_runtime.h>
#include <hip/hip_bf16.h>
#include <stdint.h>

// ---------------------------------------------------------------------------
// CLAHE for x:(8,3,1024,1024) f32, GRID=8x8, NUM_BINS=256, CLIP_LIMIT=40
// tile = 128x128 (1024 divides evenly -> no padding), pixels/tile = 16384
// clip max_val = 40*16384/256 = 2560
// ---------------------------------------------------------------------------

typedef float v2f __attribute__((ext_vector_type(2)));
typedef float v4f __attribute__((ext_vector_type(4)));
typedef float v8f __attribute__((ext_vector_type(8)));

#define NB        256
#define TH        128
#define GH        8
#define GW        8
#define IMG_H     1024
#define IMG_W     1024
#define PIX       16384
#define MAXV      2560.0f
#define LUT_SCALE (255.0f / 16384.0f)
#define NWAVE     8            // 256 threads / wave32
#define ROWSPB    8            // rows per block in apply kernel

// -------- async global->LDS helpers (CDNA5 ASYNCcnt path) ------------------
__device__ __forceinline__ void async_load_b128(unsigned lds_off, const void* gptr) {
    uint64_t ga = (uint64_t)(uintptr_t)gptr;
    asm volatile("global_load_async_to_lds_b128 %0, %1, off"
                 :: "v"(lds_off), "v"(ga) : "memory");
}
__device__ __forceinline__ void wait_asynccnt0() {
#if __has_builtin(__builtin_amdgcn_s_wait_asynccnt)
    __builtin_amdgcn_s_wait_asynccnt(0);
#else
    asm volatile("s_wait_asynccnt 0" ::: "memory");
#endif
}
// generic pointer to a __shared__ object: low 32 bits are the LDS byte offset
__device__ __forceinline__ unsigned lds_offset_of(const void* p) {
    return (unsigned)(uintptr_t)p;
}

// ===========================================================================
// Kernel 1: per-tile histogram -> clip -> redistribute -> cumsum -> LUT
// cumsum(256) done entirely with V_WMMA_F32_16X16X4_F32 (exact in fp32):
//   view hist as 16x16 matrix A;  Cum = A*U + Lstrict*T
//   U = upper-tri ones (within-row inclusive prefix)
//   T[k][*] = rowTotal[k]        (exclusive row offset, broadcast over cols)
// grid = 1536 blocks (one per tile), 256 threads (8 waves)
// ===========================================================================
__global__ void __launch_bounds__(256)
clahe_lut_kernel(const float* __restrict__ x, float* __restrict__ lut)
{
    __shared__ unsigned histW[NWAVE * NB];  // per-wave private histograms, 8KB
    __shared__ float    red[NB];
    __shared__ float    histF[NB];
    __shared__ float    rowP[NB];           // final cumsum (16x16)
    __shared__ float    rowTot[16];         // per-row totals

    const int tid = threadIdx.x;
    const int t   = blockIdx.x;             // tile id: ((b*C+c)*GH+gy)*GW+gx
    const int bc  = t >> 6;
    const int gy  = (t >> 3) & 7;
    const int gx  = t & 7;

    const float* src = x + ((size_t)bc * IMG_H + (size_t)gy * TH) * IMG_W
                         + (size_t)gx * TH;

    #pragma unroll
    for (int w = 0; w < NWAVE; ++w) histW[w * NB + tid] = 0u;
    __syncthreads();

    unsigned* myHist = &histW[(tid >> 5) * NB];

    // 16384 px = 4096 float4; 16 float4 per thread, coalesced
    #pragma unroll 4
    for (int it = 0; it < 16; ++it) {
        int id   = tid + (it << 8);         // 0..4095
        int row  = id >> 5;                 // 32 float4 per 128-px row
        int c4   = id & 31;
        v4f v = *(const v4f*)(src + (size_t)row * IMG_W + (c4 << 2));
        #pragma unroll
        for (int e = 0; e < 4; ++e) {
            int b = min(max((int)(v[e] * 256.0f), 0), 255);
            atomicAdd(&myHist[b], 1u);
        }
    }
    __syncthreads();

    // combine per-wave histograms for bin `tid`
    unsigned tot = 0;
    #pragma unroll
    for (int w = 0; w < NWAVE; ++w) tot += histW[w * NB + tid];

    // clip + reduce (all values exact small integers in f32)
    float h = fminf((float)tot, MAXV);
    red[tid] = h;
    __syncthreads();
    #pragma unroll
    for (int s = 128; s > 0; s >>= 1) {
        if (tid < s) red[tid] += red[tid + s];
        __syncthreads();
    }
    const int clipped  = PIX - (int)red[0];
    const int residual = clipped & 255;
    const float addAll = (float)((clipped - residual) >> 8);
    histF[tid] = h + addAll + ((tid < residual) ? 1.0f : 0.0f);
    __syncthreads();

    // ---- chain 1: acc = A x U  (four 16x16x4 f32 WMMAs) ----
    const int ln = tid & 15;   // A row / B col / D col
    const int hf = (tid >> 4) & 1;
    v8f acc = {};
    if (tid < 32) {                         // wave 0 only; EXEC all-ones
        #pragma unroll
        for (int q = 0; q < 4; ++q) {
            const int k0 = 4 * q + 2 * hf;  // operand k = k0 + j (j = vgpr idx)
            v2f a, b;
            a.x = histF[ln * 16 + k0];
            a.y = histF[ln * 16 + k0 + 1];
            b.x = (k0     <= ln) ? 1.0f : 0.0f;   // U[k][n] = (k <= n)
            b.y = (k0 + 1 <= ln) ? 1.0f : 0.0f;
            acc = __builtin_amdgcn_wmma_f32_16x16x4_f32(
                      false, a, false, b, (short)0, acc, false, false);
        }
        // lanes with ln==15 hold column 15 (row totals) for rows hf*8..hf*8+7
        if (ln == 15) {
            #pragma unroll
            for (int i = 0; i < 8; ++i) rowTot[i + 8 * hf] = acc[i];
        }
    }
    __syncthreads();

    // ---- chain 2: acc += Lstrict x T  (exclusive row offsets, 4 more WMMAs)
    if (tid < 32) {
        #pragma unroll
        for (int q = 0; q < 4; ++q) {
            const int k0 = 4 * q + 2 * hf;
            v2f a, b;
            a.x = (k0     < ln) ? 1.0f : 0.0f;    // Lstrict[n][k] = (k < n)
            a.y = (k0 + 1 < ln) ? 1.0f : 0.0f;
            b.x = rowTot[k0];                     // T[k][*] = rowTotal[k]
            b.y = rowTot[k0 + 1];
            acc = __builtin_amdgcn_wmma_f32_16x16x4_f32(
                      false, a, false, b, (short)0, acc, false, false);
        }
        // D layout: VGPR i -> row (i + 8*hf), col = ln
        #pragma unroll
        for (int i = 0; i < 8; ++i)
            rowP[(i + 8 * hf) * 16 + ln] = acc[i];
    }
    __syncthreads();

    float l = floorf(fminf(fmaxf(rowP[tid] * LUT_SCALE, 0.0f), 255.0f));
    lut[(size_t)t * NB + tid] = l;
}

// ===========================================================================
// Kernel 2: apply LUTs with bilinear blend.
// One block per 8-row group of one (b,c) image (r0/r1 constant per group).
// Stage the 16 needed LUTs (rows r0,r1 x 8 cols x 256 floats = 16KB) into LDS
// via async global->LDS b128; x-interp coefficients computed once per thread.
// ===========================================================================
__global__ void __launch_bounds__(256)
clahe_apply_kernel(const float* __restrict__ x, const float* __restrict__ lut,
                   float* __restrict__ out)
{
    __shared__ float ldsLUT[2 * GW * NB];   // [rowSel][gx][bin], 16KB

    const int tid = threadIdx.x;
    const int blk = blockIdx.x;
    const int bc  = blk >> 7;               // 128 groups of 8 rows per image
    const int y0  = (blk & 127) << 3;       // first row of this group

    // vertical interpolation coords (constant r0/r1 within 8-row group)
    const int jy = y0 >> 6;
    int r0 = (jy >= 1) ? ((jy - 1) >> 1) : 0;  r0 = min(r0, GH - 1);
    const int r1 = min(r0 + 1, GH - 1);

    const float* lutRow0 = lut + (((size_t)bc * GH + r0) * GW) * NB;
    const float* lutRow1 = lut + (((size_t)bc * GH + r1) * GW) * NB;

    // stage 2 x 2048 floats = 2 x 512 float4 with async loads
    const unsigned ldsBase = lds_offset_of(ldsLUT);
    #pragma unroll
    for (int k = 0; k < 2; ++k) {
        int i = tid + (k << 8);             // 0..511 (float4 units)
        async_load_b128(ldsBase + (unsigned)(i << 4),          lutRow0 + (i << 2));
        async_load_b128(ldsBase + 8192u + (unsigned)(i << 4),  lutRow1 + (i << 2));
    }

    // horizontal interp coefficients for this thread's 4 columns (row-invariant)
    int   c0e[4], c1e[4];
    float wxe[4];
    #pragma unroll
    for (int e = 0; e < 4; ++e) {
        const int pos = (tid << 2) + e;
        const int jx = pos >> 6;
        const int px = pos & 63;
        int c0 = (jx >= 1) ? ((jx - 1) >> 1) : 0;  c0 = min(c0, GW - 1);
        c0e[e] = c0;
        c1e[e] = min(c0 + 1, GW - 1);
        wxe[e] = (jx == 0) ? 1.0f
               : ((jx & 1) ? (float)(127 - px) : (float)(63 - px)) * (1.0f / 127.0f);
    }

    wait_asynccnt0();
    __syncthreads();

    const float* inBase  = x   + ((size_t)bc * IMG_H + y0) * IMG_W;
    float*       outBase = out + ((size_t)bc * IMG_H + y0) * IMG_W;

    #pragma unroll
    for (int r = 0; r < ROWSPB; ++r) {
        const int y  = y0 + r;
        const int py = y & 63;
        const float wy = (jy == 0) ? 1.0f
                       : ((jy & 1) ? (float)(127 - py) : (float)(63 - py)) * (1.0f / 127.0f);

        v4f v = *(const v4f*)(inBase + (size_t)r * IMG_W + (tid << 2));
        v4f o;
        #pragma unroll
        for (int e = 0; e < 4; ++e) {
            int idx = min(max((int)(v[e] * 255.0f), 0), 255);
            const float wx  = wxe[e];
            const float l00 = ldsLUT[c0e[e] * NB + idx];
            const float l01 = ldsLUT[c1e[e] * NB + idx];
            const float l10 = ldsLUT[2048 + c0e[e] * NB + idx];
            const float l11 = ldsLUT[2048 + c1e[e] * NB + idx];
            o[e] = (wy * (wx * l00 + (1.0f - wx) * l01)
                  + (1.0f - wy) * (wx * l10 + (1.0f - wx) * l11)) * (1.0f / 255.0f);
        }
        // nontemporal: keep the write stream out of L2 so the input stays hot
        __builtin_nontemporal_store(o, (v4f*)(outBase + (size_t)r * IMG_W + (tid << 2)));
    }
}

// ===========================================================================
extern "C" void kernel_launch(void* const* d_in, const int* in_sizes, int n_in,
                              void* d_out, int out_size, void* d_ws, size_t ws_size,
                              hipStream_t stream)
{
    const float* x   = (const float*)d_in[0];
    float*       out = (float*)d_out;
    float*       lut = (float*)d_ws;   // 1536 * 256 floats = 1.5 MB scratch

    // one block per tile: 8*3*8*8 = 1536
    clahe_lut_kernel<<<1536, 256, 0, stream>>>(x, lut);
    // one block per 8-row group: 8*3*1024/8 = 3072
    clahe_apply_kernel<<<3072, 256, 0, stream>>>(x, lut, out);
}